// TopKSoftmaxGate_pytorch_69037304316406
// MI455X (gfx1250) — compile-verified
//
#include <hip/hip_runtime.h>
#include <hip/hip_bf16.h>

// ---------------------------------------------------------------------------
// y[b,d] = sum_e h[b,d,e] * gate[e], gate = sparse top-2 softmax of
// (expert_weights + bias) over E=16 experts.
//
// Memory-bound: ~403 MB read of h + 25 MB write -> ~18.4 us floor @ 23.3 TB/s.
// Strategy: one streaming pass over h with per-lane global_load_b64 that maps
// directly onto the V_WMMA_F32_16X16X4_F32 A-matrix layout; 4 chained WMMAs
// reduce the E=16 axis per 16-row tile. Non-temporal hints since h (403MB)
// overflows the 192MB L2 and is touched exactly once.
// ---------------------------------------------------------------------------

typedef float v2f __attribute__((ext_vector_type(2)));
typedef float v8f __attribute__((ext_vector_type(8)));

// ---- Kernel 1: compute the 16-element sparse gate into d_ws ---------------
__global__ void gate_kernel(const float* __restrict__ expert_weights,
                            const float* __restrict__ bias,
                            float* __restrict__ gate, int E) {
  if (threadIdx.x == 0 && blockIdx.x == 0) {
    // logits[e] = expert_weights[e] + bias[e]
    // top-2 with lax.top_k tie-breaking (first/lowest index wins on ties)
    float best = -3.402823466e+38f; int bi = 0;
    for (int e = 0; e < E; ++e) {
      float v = expert_weights[e] + bias[e];
      if (v > best) { best = v; bi = e; }
    }
    float second = -3.402823466e+38f; int si = 0;
    for (int e = 0; e < E; ++e) {
      if (e == bi) continue;
      float v = expert_weights[e] + bias[e];
      if (v > second) { second = v; si = e; }
    }
    // softmax over [best, second]; second <= best so exp arg <= 0 (stable)
    float e1 = __expf(second - best);
    float inv = 1.0f / (1.0f + e1);
    for (int e = 0; e < E; ++e) gate[e] = 0.0f;
    gate[bi] = inv;
    gate[si] = e1 * inv;
  }
}

// ---- Kernel 2: WMMA combine ------------------------------------------------
// One 16-row tile (16 rows x 16 experts, f32) per wave.
// A-matrix 16x4 f32 layout: lanes 0-15 hold M=lane, K={0,1} in vgpr{0,1};
// lanes 16-31 hold M=lane-16, K={2,3}. -> per-lane float2 load at
// row*16 + chunk*4 + (lane>=16)*2. B is the gate broadcast across N.
// D: acc[v] at lanes with (lane&15)==0 holds y[row v + 8*(lane>>4)].
__global__ void __launch_bounds__(256)
combine_wmma(const float* __restrict__ h,
             const float* __restrict__ gate,
             float* __restrict__ out,
             int numTiles) {
  __shared__ float lds[8 * 16];  // 8 waves/block * 16 rows

  const int lane       = threadIdx.x & 31;
  const int waveInBlk  = threadIdx.x >> 5;
  const int tile       = blockIdx.x * 8 + waveInBlk;  // wave-uniform
  if (tile >= numTiles) return;  // uniform exit -> EXEC stays all-1s for WMMA

  const int rowInTile = lane & 15;
  const int kHalf     = lane >> 4;          // 0: K={0,1}, 1: K={2,3}
  const long long r0  = (long long)tile * 16;

  const float* rowPtr = h + (r0 + rowInTile) * 16 + kHalf * 2;

  // B operand: gate values; K->(vgpr,lane-half) mapping mirrors A.
  v2f b0 = *(const v2f*)(gate + 0 + kHalf * 2);
  v2f b1 = *(const v2f*)(gate + 4 + kHalf * 2);
  v2f b2 = *(const v2f*)(gate + 8 + kHalf * 2);
  v2f b3 = *(const v2f*)(gate + 12 + kHalf * 2);

  // A operand: 4 x global_load_b64, wave covers 1KB contiguous, non-temporal.
  v2f a0 = __builtin_nontemporal_load((const v2f*)(rowPtr + 0));
  v2f a1 = __builtin_nontemporal_load((const v2f*)(rowPtr + 4));
  v2f a2 = __builtin_nontemporal_load((const v2f*)(rowPtr + 8));
  v2f a3 = __builtin_nontemporal_load((const v2f*)(rowPtr + 12));

  v8f acc = {};
  // 8 args: (neg_a, A, neg_b, B, c_mod, C, reuse_a, reuse_b)
  acc = __builtin_amdgcn_wmma_f32_16x16x4_f32(false, a0, false, b0, (short)0, acc, false, false);
  acc = __builtin_amdgcn_wmma_f32_16x16x4_f32(false, a1, false, b1, (short)0, acc, false, false);
  acc = __builtin_amdgcn_wmma_f32_16x16x4_f32(false, a2, false, b2, (short)0, acc, false, false);
  acc = __builtin_amdgcn_wmma_f32_16x16x4_f32(false, a3, false, b3, (short)0, acc, false, false);

  // Extract column N=0: lanes 0 and 16 hold y for rows {v, v+8} in acc[v].
  // Same-wave LDS bounce (DS ops are in-order within a wave; no barrier).
  float* wlds = lds + waveInBlk * 16;
  if (rowInTile == 0) {
#pragma unroll
    for (int v = 0; v < 8; ++v) wlds[v + kHalf * 8] = acc[v];
  }
  if (lane < 16) {
    __builtin_nontemporal_store(wlds[lane], out + r0 + lane);
  }
}

// ---------------------------------------------------------------------------
extern "C" void kernel_launch(void* const* d_in, const int* in_sizes, int n_in,
                              void* d_out, int out_size, void* d_ws, size_t ws_size,
                              hipStream_t stream) {
  const float* h    = (const float*)d_in[0];  // [B, D, E]
  // d_in[1] = x (unused), d_in[2] = permutation_weights (unused)
  const float* ew   = (const float*)d_in[3];  // [E, 1]
  const float* bias = (const float*)d_in[4];  // [E]
  float* out  = (float*)d_out;                // [B, D]
  float* gate = (float*)d_ws;                 // 16 floats of scratch

  const int E = in_sizes[3];                       // 16
  const long long R = (long long)in_sizes[0] / E;  // B*D = 6,291,456
  const int numTiles = (int)(R / 16);              // 393,216 (exact)

  gate_kernel<<<1, 32, 0, stream>>>(ew, bias, gate, E);

  const int wavesPerBlock = 8;  // 256 threads = 8 wave32
  const int blocks = (numTiles + wavesPerBlock - 1) / wavesPerBlock;
  combine_wmma<<<blocks, 256, 0, stream>>>(h, gate, out, numTiles);
}